// GraphFlixModel_25185688224059
// MI455X (gfx1250) — compile-verified
//
#include <hip/hip_runtime.h>

// ---------------------------------------------------------------------------
// HGT (2-layer, user/movie bipartite) for MI455X / gfx1250.
// Dense GEMMs: V_WMMA_F32_16X16X4_F32 (fp32 matrix core, wave32), with the
// weight matrix staged into LDS in WMMA-fragment order (1 ds_load_b64 per
// B fragment) and one wave producing a 16x128 output strip (8x A reuse).
// Edge phase (segment softmax + scatter) uses float atomics; working set
// (~92 MB of node features) fits the 192 MB L2, so gathers run at L2 BW.
// ---------------------------------------------------------------------------

typedef __attribute__((ext_vector_type(2))) float v2f;
typedef __attribute__((ext_vector_type(8))) float v8f;

#define NUQ   50000
#define NMQ   10000
#define NEQ   500000
#define CDIM  128
#define HH    4
#define DDIM  32
#define DPHIQ 64
#define LLQ   2

__device__ __forceinline__ float gelu_f(float x) {
  float x3 = x * x * x;
  return 0.5f * x * (1.0f + tanhf(0.7978845608028654f * (x + 0.044715f * x3)));
}
// Order-preserving float<->uint encoding for atomicMax over signed floats.
__device__ __forceinline__ unsigned enc_f(float f) {
  unsigned u = __float_as_uint(f);
  return (u & 0x80000000u) ? ~u : (u | 0x80000000u);
}
__device__ __forceinline__ float dec_f(unsigned u) {
  unsigned v = (u & 0x80000000u) ? (u ^ 0x80000000u) : ~u;
  return __uint_as_float(v);
}

// ---------------------------------------------------------------------------
// Y[n x 128] = act(X[n x K]) @ W[K x 128]  (+ epilogue)
// AMODE: 0 identity, 1 gelu(X)
// EPI:   0  Y = acc + bias
//        1  Y = resid + acc + bias
//        2  g = sigmoid(*skipp); Y = g*(acc+bias) + (1-g)*resid
//
// One wave per 16-row strip, all 8 N-tiles (8 v8f accumulators).
// W is cooperatively staged into LDS pre-swizzled into WMMA B-fragment order
//   wsh[((kstep*8 + tn)*32 + lane)*2 + v] = W[kstep*4 + 2*(lane>>4) + v][tn*16 + (lane&15)]
// so the inner loop does: 1 global_load_b64 (A) + 8 ds_load_b64 (B) + 8 wmma.
//
// WMMA f32 16x16x4 fragment layout:
//   A(16x4):  lane = (k>=2?16:0)+m, vgpr v -> A[m][2*(lane>>4)+v]
//   B(4x16):  lane = (k>=2?16:0)+n, vgpr v -> B[2*(lane>>4)+v][n]
//   C/D:      vgpr i, lanes 0-15 -> (M=i, N=lane); lanes 16-31 -> (M=i+8)
// ---------------------------------------------------------------------------
template <int AMODE, int EPI>
__global__ __launch_bounds__(256) void gemm_n128(
    const float* __restrict__ X, const float* __restrict__ W,
    const float* __restrict__ bias, const float* __restrict__ resid,
    const float* __restrict__ skipp, float* __restrict__ Y, int n, int K) {
  __shared__ float wsh[16384];  // 64 KB: up to 32 ksteps x 8 tn x 32 lanes x 2
  const int tid = threadIdx.x;
  const int ksteps = K >> 2;

  // Cooperative swizzled fill of W into fragment order.
  for (int i = tid; i < ksteps * 256; i += 256) {
    const int kstep = i >> 8;
    const int rem = i & 255;
    const int tn = rem >> 5;
    const int ln = rem & 31;
    const int krow = kstep * 4 + 2 * (ln >> 4);
    const int col = tn * 16 + (ln & 15);
    wsh[2 * i]     = W[(size_t)krow * 128 + col];
    wsh[2 * i + 1] = W[(size_t)(krow + 1) * 128 + col];
  }
  __syncthreads();

  const int wave = (blockIdx.x * blockDim.x + tid) >> 5;
  const int lane = tid & 31;
  const int tilesM = n >> 4;
  if (wave >= tilesM) return;  // wave-uniform: EXEC stays all-ones for WMMA
  const int m = lane & 15;
  const int half = lane >> 4;

  const float* ap = X + (size_t)(wave * 16 + m) * K + 2 * half;
  __builtin_prefetch(ap + 64, 0, 1);  // global_prefetch_b8: pull strip into caches
  v8f acc[8] = {};
  for (int ks = 0; ks < ksteps; ++ks) {
    v2f a = *(const v2f*)ap;  // 8B-aligned: K, ks multiples of 4
    if (AMODE == 1) { a[0] = gelu_f(a[0]); a[1] = gelu_f(a[1]); }
#pragma unroll
    for (int tn = 0; tn < 8; ++tn) {
      const v2f b = *(const v2f*)(wsh + (((ks * 8 + tn) * 32 + lane) << 1));
      acc[tn] = __builtin_amdgcn_wmma_f32_16x16x4_f32(false, a, false, b, (short)0,
                                                      acc[tn], false, false);
    }
    ap += 4;
  }

  float g = 1.0f;
  if (EPI == 2) g = 1.0f / (1.0f + __expf(-skipp[0]));
#pragma unroll
  for (int tn = 0; tn < 8; ++tn) {
    const int col = tn * 16 + m;
    const float bv = bias ? bias[col] : 0.0f;
#pragma unroll
    for (int i = 0; i < 8; ++i) {
      const int row = wave * 16 + i + 8 * half;
      float v = acc[tn][i] + bv;
      if (EPI == 1) v += resid[(size_t)row * 128 + col];
      if (EPI == 2) v = g * v + (1.0f - g) * resid[(size_t)row * 128 + col];
      Y[(size_t)row * 128 + col] = v;
    }
  }
}

// ---------------------------------------------------------------------------
// Per-head relation transform: Y[n, h*32+e] = scale_h * sum_d X[n,h*32+d]*A[h,d,e]
// scale_h = p_rel[h]/sqrt(32) when prel != null (folds logit scaling into ke).
// One wave per (row-tile, head): computes both 16-wide N-tiles, A loaded once.
// ---------------------------------------------------------------------------
__global__ __launch_bounds__(256) void rel_kernel(const float* __restrict__ X,
                                                  const float* __restrict__ A,
                                                  const float* __restrict__ prel,
                                                  float* __restrict__ Y, int n) {
  const int wave = (blockIdx.x * blockDim.x + threadIdx.x) >> 5;
  const int lane = threadIdx.x & 31;
  const int tilesM = n >> 4;
  if (wave >= tilesM * HH) return;
  const int tm = wave >> 2;
  const int h = wave & 3;
  const int m = lane & 15;
  const int half = lane >> 4;

  const float* ap = X + (size_t)(tm * 16 + m) * 128 + h * 32 + 2 * half;
  const float* bp = A + h * 1024 + (size_t)(2 * half) * 32 + m;
  v8f acc0 = {}, acc1 = {};
#pragma unroll
  for (int k0 = 0; k0 < 32; k0 += 4) {
    v2f a = *(const v2f*)ap;
    v2f b0, b1;
    b0[0] = bp[0];  b0[1] = bp[32];
    b1[0] = bp[16]; b1[1] = bp[48];
    acc0 = __builtin_amdgcn_wmma_f32_16x16x4_f32(false, a, false, b0, (short)0, acc0,
                                                 false, false);
    acc1 = __builtin_amdgcn_wmma_f32_16x16x4_f32(false, a, false, b1, (short)0, acc1,
                                                 false, false);
    ap += 4;
    bp += 128;
  }
  const float scale = prel ? prel[h] * 0.17677669529663687f : 1.0f;  // 1/sqrt(32)
#pragma unroll
  for (int i = 0; i < 8; ++i) {
    const int row = tm * 16 + i + 8 * half;
    Y[(size_t)row * 128 + h * 32 + m]      = acc0[i] * scale;
    Y[(size_t)row * 128 + h * 32 + 16 + m] = acc1[i] * scale;
  }
}

// ---------------------------------------------------------------------------
__global__ void fill_u32(unsigned* __restrict__ p, unsigned v, int n) {
  int t = blockIdx.x * blockDim.x + threadIdx.x;
  if (t < n) p[t] = v;
}

// logits[e,h] = dot(q[dst,h,:], ke[src,h,:])  (scale already folded into ke)
__global__ void logits_kernel(const float* __restrict__ Q, const float* __restrict__ KE,
                              const int* __restrict__ src, const int* __restrict__ dst,
                              float* __restrict__ logits, unsigned* __restrict__ mbuf) {
  int t = blockIdx.x * blockDim.x + threadIdx.x;
  if (t >= NEQ * HH) return;
  int e = t >> 2, h = t & 3;
  int si = src[e], di = dst[e];
  const float4* qp = (const float4*)(Q + (size_t)di * CDIM + h * DDIM);
  const float4* kp = (const float4*)(KE + (size_t)si * CDIM + h * DDIM);
  float acc = 0.0f;
#pragma unroll
  for (int i = 0; i < 8; ++i) {
    float4 a = qp[i], b = kp[i];
    acc += a.x * b.x + a.y * b.y + a.z * b.z + a.w * b.w;
  }
  logits[t] = acc;
  atomicMax(mbuf + (size_t)di * HH + h, enc_f(acc));
}

// logits[e,h] <- exp(logits - max[dst]);  s[dst] += exp(...)
__global__ void expsum_kernel(float* __restrict__ logits, const unsigned* __restrict__ mbuf,
                              float* __restrict__ sbuf, const int* __restrict__ dst) {
  int t = blockIdx.x * blockDim.x + threadIdx.x;
  if (t >= NEQ * HH) return;
  int e = t >> 2, h = t & 3;
  int di = dst[e];
  float mv = dec_f(mbuf[(size_t)di * HH + h]);
  float ev = __expf(logits[t] - mv);
  logits[t] = ev;
  atomicAdd(sbuf + (size_t)di * HH + h, ev);
}

// agg[dst, h*32+c] += (e / (s[dst]+1e-16)) * ve[src, h*32+c]
__global__ void scatter_kernel(const float* __restrict__ logits, const float* __restrict__ sbuf,
                               const float* __restrict__ VE, const int* __restrict__ src,
                               const int* __restrict__ dst, float* __restrict__ AGG) {
  int t = blockIdx.x * blockDim.x + threadIdx.x;
  if (t >= NEQ * HH) return;
  int e = t >> 2, h = t & 3;
  int si = src[e], di = dst[e];
  float alpha = logits[t] / (sbuf[(size_t)di * HH + h] + 1e-16f);
  const float4* vp = (const float4*)(VE + (size_t)si * CDIM + h * DDIM);
  float* ap = AGG + (size_t)di * CDIM + h * DDIM;
#pragma unroll
  for (int i = 0; i < 8; ++i) {
    float4 v = vp[i];
    atomicAdd(ap + 4 * i + 0, alpha * v.x);
    atomicAdd(ap + 4 * i + 1, alpha * v.y);
    atomicAdd(ap + 4 * i + 2, alpha * v.z);
    atomicAdd(ap + 4 * i + 3, alpha * v.w);
  }
}

// ---------------------------------------------------------------------------
extern "C" void kernel_launch(void* const* d_in, const int* in_sizes, int n_in,
                              void* d_out, int out_size, void* d_ws, size_t ws_size,
                              hipStream_t stream) {
  const float* user_emb  = (const float*)d_in[0];
  const float* movie_emb = (const float*)d_in[1];
  const float* phi       = (const float*)d_in[2];
  const float* meta_w    = (const float*)d_in[3];
  const float* meta_b    = (const float*)d_in[4];
  const float* kw        = (const float*)d_in[5];
  const float* kb        = (const float*)d_in[6];
  const float* qw        = (const float*)d_in[7];
  const float* qb        = (const float*)d_in[8];
  const float* vw        = (const float*)d_in[9];
  const float* vb        = (const float*)d_in[10];
  const float* a_rel     = (const float*)d_in[11];
  const float* m_rel     = (const float*)d_in[12];
  const float* p_rel     = (const float*)d_in[13];
  const float* ow        = (const float*)d_in[14];
  const float* ob        = (const float*)d_in[15];
  const float* skip      = (const float*)d_in[16];
  const int*   ei        = (const int*)d_in[17];  // [2,E]: row0 user, row1 movie

  // Node features live directly in d_out: [xu (NU*128) | xm (NM*128)]
  float* xu = (float*)d_out;
  float* xm = xu + (size_t)NUQ * CDIM;

  // Workspace layout (floats). Total ~45.9M floats (~184 MB).
  float* W = (float*)d_ws;
  const size_t NT = (size_t)NUQ + NMQ;
  float* Kb  = W;
  float* Qb  = Kb + NT * CDIM;
  float* Vb  = Qb + NT * CDIM;
  float* KE  = Vb + NT * CDIM;
  float* VE  = KE + (size_t)NUQ * CDIM;
  float* LG  = VE + (size_t)NUQ * CDIM;
  float* Mb  = LG + (size_t)NEQ * HH;     // encoded-uint segment max
  float* Sb  = Mb + (size_t)NUQ * HH;     // segment sum
  float* AGU = Sb + (size_t)NUQ * HH;
  float* AGM = AGU + (size_t)NUQ * CDIM;

  const int ns[2] = {NUQ, NMQ};
  float* xs[2] = {xu, xm};
  float* kk[2] = {Kb, Kb + (size_t)NUQ * CDIM};
  float* qq[2] = {Qb, Qb + (size_t)NUQ * CDIM};
  float* vv[2] = {Vb, Vb + (size_t)NUQ * CDIM};
  float* ag[2] = {AGU, AGM};

  const int THREADS = 256;  // 8 wave32 waves per workgroup
  const int NEH = NEQ * HH;
  const int edgeBlocks = (NEH + THREADS - 1) / THREADS;
  // Strip GEMM: one wave per 16-row strip -> blocks = ceil((n/16)/8)
  auto gemmBlocks = [](int n) { return ((n >> 4) + 7) / 8; };
  // rel: one wave per (row-tile, head) -> blocks = ceil((n/16)*4/8)
  auto relBlocks = [](int n) { return ((n >> 4) * HH + 7) / 8; };

  // xu = user_emb
  hipMemcpyAsync(xu, user_emb, (size_t)NUQ * CDIM * sizeof(float),
                 hipMemcpyDeviceToDevice, stream);
  // xm = movie_emb + phi @ meta_w + meta_b
  gemm_n128<0, 1><<<gemmBlocks(NMQ), THREADS, 0, stream>>>(phi, meta_w, meta_b,
                                                           movie_emb, nullptr, xm,
                                                           NMQ, DPHIQ);

  for (int l = 0; l < LLQ; ++l) {
    // --- K/Q/V projections (WMMA GEMMs) ---
    for (int t = 0; t < 2; ++t) {
      const size_t wo = (size_t)(l * 2 + t) * CDIM * CDIM;
      const size_t bo = (size_t)(l * 2 + t) * CDIM;
      gemm_n128<0, 0><<<gemmBlocks(ns[t]), THREADS, 0, stream>>>(
          xs[t], kw + wo, kb + bo, nullptr, nullptr, kk[t], ns[t], CDIM);
      gemm_n128<0, 0><<<gemmBlocks(ns[t]), THREADS, 0, stream>>>(
          xs[t], qw + wo, qb + bo, nullptr, nullptr, qq[t], ns[t], CDIM);
      gemm_n128<0, 0><<<gemmBlocks(ns[t]), THREADS, 0, stream>>>(
          xs[t], vw + wo, vb + bo, nullptr, nullptr, vv[t], ns[t], CDIM);
    }
    // --- Edge types: r=0 user->movie, r=1 movie->user ---
    for (int r = 0; r < 2; ++r) {
      const int st = (r == 0) ? 0 : 1;
      const int dt = 1 - st;
      const int* srcp = (r == 0) ? ei : ei + NEQ;
      const int* dstp = (r == 0) ? ei + NEQ : ei;
      const size_t ro = (size_t)(l * 2 + r) * HH * DDIM * DDIM;

      rel_kernel<<<relBlocks(ns[st]), THREADS, 0, stream>>>(
          kk[st], a_rel + ro, p_rel + (l * 2 + r) * HH, KE, ns[st]);
      rel_kernel<<<relBlocks(ns[st]), THREADS, 0, stream>>>(vv[st], m_rel + ro,
                                                            nullptr, VE, ns[st]);
      const int ndh = ns[dt] * HH;
      fill_u32<<<(ndh + THREADS - 1) / THREADS, THREADS, 0, stream>>>(
          (unsigned*)Mb, 0x007FFFFFu /* enc(-inf) */, ndh);
      fill_u32<<<(ndh + THREADS - 1) / THREADS, THREADS, 0, stream>>>((unsigned*)Sb, 0u,
                                                                      ndh);
      const int ndc = ns[dt] * CDIM;
      fill_u32<<<(ndc + THREADS - 1) / THREADS, THREADS, 0, stream>>>(
          (unsigned*)ag[dt], 0u, ndc);

      logits_kernel<<<edgeBlocks, THREADS, 0, stream>>>(qq[dt], KE, srcp, dstp, LG,
                                                        (unsigned*)Mb);
      expsum_kernel<<<edgeBlocks, THREADS, 0, stream>>>(LG, (unsigned*)Mb, Sb, dstp);
      scatter_kernel<<<edgeBlocks, THREADS, 0, stream>>>(LG, Sb, VE, srcp, dstp,
                                                         ag[dt]);
    }
    // --- Output projection: x = g*(gelu(agg)@ow + ob) + (1-g)*x (in place) ---
    for (int t = 0; t < 2; ++t) {
      const size_t wo = (size_t)(l * 2 + t) * CDIM * CDIM;
      const size_t bo = (size_t)(l * 2 + t) * CDIM;
      gemm_n128<1, 2><<<gemmBlocks(ns[t]), THREADS, 0, stream>>>(
          ag[t], ow + wo, ob + bo, xs[t], skip + (l * 2 + t), xs[t], ns[t], CDIM);
    }
  }
}